// GCN_73203422593427
// MI455X (gfx1250) — compile-verified
//
#include <hip/hip_runtime.h>
#include <math.h>

// ---- problem constants (match reference) ------------------------------------
#define N_NODES    100000
#define N_EDGES    1600000
#define HIDDEN     64
#define NUM_FEAT   5
#define NUM_CLS    5

typedef __attribute__((ext_vector_type(2))) float v2f;
typedef __attribute__((ext_vector_type(8))) float v8f;

// ---- small helpers ----------------------------------------------------------
__global__ void gcn_zero_i32(int* __restrict__ p, int n) {
    int i = blockIdx.x * blockDim.x + threadIdx.x;
    if (i < n) p[i] = 0;
}

// degc[dst] += 1 over all edges (int histogram; feeds both dinv and CSR rowptr)
__global__ void gcn_deg_count(const int* __restrict__ dst, int* __restrict__ degc) {
    int e = blockIdx.x * blockDim.x + threadIdx.x;
    if (e < N_EDGES) atomicAdd(&degc[dst[e]], 1);
}

// dinv[i] = rsqrt(deg + 1)  (self loop)
__global__ void gcn_dinv(const int* __restrict__ degc, float* __restrict__ dinv) {
    int i = blockIdx.x * blockDim.x + threadIdx.x;
    if (i < N_NODES) dinv[i] = rsqrtf((float)degc[i] + 1.0f);
}

// exclusive prefix scan of degc -> rowptr[N+1]; single 1024-thread block,
// Hillis-Steele per 1024-chunk with running carry. ~100 chunks, trivial cost.
__global__ __launch_bounds__(1024)
void gcn_scan(const int* __restrict__ degc, int* __restrict__ rowptr) {
    __shared__ int sm[1024];
    __shared__ int carry;
    const int tid = threadIdx.x;
    if (tid == 0) carry = 0;
    __syncthreads();
    for (int base = 0; base < N_NODES; base += 1024) {
        int i = base + tid;
        int v = (i < N_NODES) ? degc[i] : 0;
        sm[tid] = v;
        __syncthreads();
#pragma unroll
        for (int off = 1; off < 1024; off <<= 1) {
            int t = (tid >= off) ? sm[tid - off] : 0;
            __syncthreads();
            sm[tid] += t;
            __syncthreads();
        }
        if (i < N_NODES) rowptr[i] = carry + (sm[tid] - v);   // exclusive
        __syncthreads();
        if (tid == 0) carry += sm[1023];
        __syncthreads();
    }
    if (tid == 0) rowptr[N_NODES] = carry;                    // == N_EDGES
}

__global__ void gcn_copy_i32(const int* __restrict__ a, int* __restrict__ b, int n) {
    int i = blockIdx.x * blockDim.x + threadIdx.x;
    if (i < n) b[i] = a[i];
}

// CSR fill: slot via per-dst cursor; value = GCN edge norm dinv[s]*dinv[d]
__global__ void gcn_csr_fill(const int* __restrict__ src, const int* __restrict__ dst,
                             const float* __restrict__ dinv, int* __restrict__ cursor,
                             int* __restrict__ col, float* __restrict__ val) {
    int e = blockIdx.x * blockDim.x + threadIdx.x;
    if (e >= N_EDGES) return;
    int s = src[e], d = dst[e];
    int pos = atomicAdd(&cursor[d], 1);
    col[pos] = s;
    val[pos] = dinv[s] * dinv[d];
}

// ---- layer-1 GEMM: h1[n,f] = sum_k x[n,3,k] * W1[k,f]   (K=5, tiny) ---------
__global__ void gcn_gemm_in(const float* __restrict__ x, const float* __restrict__ W1,
                            float* __restrict__ h) {
    int idx = blockIdx.x * blockDim.x + threadIdx.x;          // N*64 threads
    if (idx >= N_NODES * HIDDEN) return;
    int n = idx >> 6, f = idx & 63;
    const float* xr = x + n * (4 * NUM_FEAT) + 3 * NUM_FEAT;  // x[n, -1, :]
    float acc = 0.0f;
#pragma unroll
    for (int k = 0; k < NUM_FEAT; ++k) acc = fmaf(xr[k], W1[k * HIDDEN + f], acc);
    h[idx] = acc;
}

// ---- layer-2 GEMM via fp32 WMMA: out = A[N,64] @ W[64,64] -------------------
// One wave per 16-row M tile; 4 v8f accumulators cover all 64 output columns.
// W staged in LDS (16 KB). K=64 -> 16 x V_WMMA_F32_16X16X4_F32 per tile.
__global__ __launch_bounds__(256)
void gcn_gemm_wmma(const float* __restrict__ A, const float* __restrict__ W,
                   float* __restrict__ out) {
    __shared__ float sW[HIDDEN * HIDDEN];                     // 64x64 = 16 KB
#pragma unroll
    for (int i = 0; i < (HIDDEN * HIDDEN) / (256 * 4); ++i)   // b128 cooperative fill
        ((float4*)sW)[threadIdx.x + i * 256] = ((const float4*)W)[threadIdx.x + i * 256];
    __syncthreads();

    const int wave = threadIdx.x >> 5;
    const int lane = threadIdx.x & 31;
    const int half = lane >> 4;            // 0: K pair {0,1}; 1: K pair {2,3}
    const int l15  = lane & 15;
    const int tile = blockIdx.x * 8 + wave;                   // 16-row tile id
    if (tile >= N_NODES / 16) return;                         // 100000 = 16*6250
    const int row0 = tile * 16;

    v8f acc0 = {}, acc1 = {}, acc2 = {}, acc3 = {};
    const float* arow = A + (row0 + l15) * HIDDEN + half * 2;

#pragma unroll
    for (int k4 = 0; k4 < HIDDEN; k4 += 4) {
        // A 16x4 tile: lane l15 = row M, VGPR0/1 = K = k4 + half*2 + {0,1}
        v2f a; a.x = arow[k4 + 0]; a.y = arow[k4 + 1];
        const int kk = k4 + half * 2;
        const float* wr0 = sW + (kk + 0) * HIDDEN + l15;      // B row K=kk,   col N=l15
        const float* wr1 = sW + (kk + 1) * HIDDEN + l15;      // B row K=kk+1
        v2f b0, b1, b2, b3;
        b0.x = wr0[0];  b0.y = wr1[0];
        b1.x = wr0[16]; b1.y = wr1[16];
        b2.x = wr0[32]; b2.y = wr1[32];
        b3.x = wr0[48]; b3.y = wr1[48];
        acc0 = __builtin_amdgcn_wmma_f32_16x16x4_f32(false, a, false, b0, (short)0, acc0, false, false);
        acc1 = __builtin_amdgcn_wmma_f32_16x16x4_f32(false, a, false, b1, (short)0, acc1, false, false);
        acc2 = __builtin_amdgcn_wmma_f32_16x16x4_f32(false, a, false, b2, (short)0, acc2, false, false);
        acc3 = __builtin_amdgcn_wmma_f32_16x16x4_f32(false, a, false, b3, (short)0, acc3, false, false);
    }

    // C/D layout: VGPR r -> M = r (lanes 0-15) or M = 8+r (lanes 16-31), N = l15
#pragma unroll
    for (int r = 0; r < 8; ++r) {
        int gr = row0 + (half ? (8 + r) : r);
        float* o = out + gr * HIDDEN + l15;
        o[0]  = acc0[r];
        o[16] = acc1[r];
        o[32] = acc2[r];
        o[48] = acc3[r];
    }
}

// ---- fused gather + self-loop + bias + ReLU (no atomics) --------------------
// out[n,f] = relu( sum_{j in in-edges(n)} h[col[j],f]*val[j] + h[n,f]*dinv[n]^2 + b[f] )
// 16 threads per node, float4 per thread -> b128 gathers, coalesced stores.
__global__ void gcn_gather(const int* __restrict__ rowptr, const int* __restrict__ col,
                           const float* __restrict__ val, const float* __restrict__ h,
                           const float* __restrict__ dinv, const float* __restrict__ b,
                           float* __restrict__ out) {
    int idx = blockIdx.x * blockDim.x + threadIdx.x;          // N*16 threads
    if (idx >= N_NODES * 16) return;
    int n  = idx >> 4;
    int f4 = (idx & 15) * 4;
    int beg = rowptr[n], end = rowptr[n + 1];
    float4 acc = make_float4(0.f, 0.f, 0.f, 0.f);
    for (int j = beg; j < end; ++j) {
        float w = val[j];
        const float4 hv = *(const float4*)(h + col[j] * HIDDEN + f4);
        acc.x = fmaf(hv.x, w, acc.x);
        acc.y = fmaf(hv.y, w, acc.y);
        acc.z = fmaf(hv.z, w, acc.z);
        acc.w = fmaf(hv.w, w, acc.w);
    }
    float di = dinv[n];
    float sn = di * di;                                       // self-loop weight 1/deg
    const float4 hs = *(const float4*)(h + n * HIDDEN + f4);
    const float4 bb = *(const float4*)(b + f4);
    float4 v;
    v.x = fmaxf(fmaf(hs.x, sn, acc.x) + bb.x, 0.f);
    v.y = fmaxf(fmaf(hs.y, sn, acc.y) + bb.y, 0.f);
    v.z = fmaxf(fmaf(hs.z, sn, acc.z) + bb.z, 0.f);
    v.w = fmaxf(fmaf(hs.w, sn, acc.w) + bb.w, 0.f);
    *(float4*)(out + n * HIDDEN + f4) = v;
}

// ---- head: softmax(h @ Wout + bout) -----------------------------------------
__global__ void gcn_head(const float* __restrict__ h, const float* __restrict__ Wout,
                         const float* __restrict__ bout, float* __restrict__ out) {
    int n = blockIdx.x * blockDim.x + threadIdx.x;
    if (n >= N_NODES) return;
    const float* hr = h + n * HIDDEN;
    float acc[NUM_CLS];
#pragma unroll
    for (int c = 0; c < NUM_CLS; ++c) acc[c] = bout[c];
    for (int k = 0; k < HIDDEN; ++k) {
        float hv = hr[k];
#pragma unroll
        for (int c = 0; c < NUM_CLS; ++c) acc[c] = fmaf(hv, Wout[k * NUM_CLS + c], acc[c]);
    }
    float m = acc[0];
#pragma unroll
    for (int c = 1; c < NUM_CLS; ++c) m = fmaxf(m, acc[c]);
    float s = 0.0f;
#pragma unroll
    for (int c = 0; c < NUM_CLS; ++c) { acc[c] = __expf(acc[c] - m); s += acc[c]; }
    float inv = 1.0f / s;
#pragma unroll
    for (int c = 0; c < NUM_CLS; ++c) out[n * NUM_CLS + c] = acc[c] * inv;
}

// ---- host orchestration -----------------------------------------------------
extern "C" void kernel_launch(void* const* d_in, const int* in_sizes, int n_in,
                              void* d_out, int out_size, void* d_ws, size_t ws_size,
                              hipStream_t stream) {
    const float* x    = (const float*)d_in[0];      // [N,4,5]
    const int*   ei   = (const int*)  d_in[1];      // [2,E] (int32 per harness)
    /* batch d_in[2] unused */
    const float* W1   = (const float*)d_in[3];
    const float* b1   = (const float*)d_in[4];
    const float* W2   = (const float*)d_in[5];
    const float* b2   = (const float*)d_in[6];
    const float* Wout = (const float*)d_in[7];
    const float* bout = (const float*)d_in[8];
    float* outp = (float*)d_out;

    const int* src = ei;
    const int* dst = ei + N_EDGES;

    // workspace layout
    char* ws = (char*)d_ws;
    int*   degc   = (int*)ws;                 ws += sizeof(int)   * N_NODES;
    int*   rowptr = (int*)ws;                 ws += sizeof(int)   * (N_NODES + 1);
    int*   cursor = (int*)ws;                 ws += sizeof(int)   * N_NODES;
    int*   col    = (int*)ws;                 ws += sizeof(int)   * N_EDGES;
    float* dinv   = (float*)ws;               ws += sizeof(float) * N_NODES;
    float* val    = (float*)ws;               ws += sizeof(float) * N_EDGES;
    float* bufA   = (float*)ws;               ws += sizeof(float) * N_NODES * HIDDEN;
    float* bufB   = (float*)ws;               /* N*64 */

    const int T = 256;
    const int gN    = (N_NODES + T - 1) / T;
    const int gE    = (N_EDGES + T - 1) / T;
    const int gNH   = (N_NODES * HIDDEN + T - 1) / T;
    const int gGat  = (N_NODES * 16 + T - 1) / T;
    const int gWmma = ((N_NODES / 16) + 7) / 8;

    // CSR build: degree -> dinv / rowptr -> bucket fill with edge norms
    gcn_zero_i32<<<gN, T, 0, stream>>>(degc, N_NODES);
    gcn_deg_count<<<gE, T, 0, stream>>>(dst, degc);
    gcn_dinv<<<gN, T, 0, stream>>>(degc, dinv);
    gcn_scan<<<1, 1024, 0, stream>>>(degc, rowptr);
    gcn_copy_i32<<<gN, T, 0, stream>>>(rowptr, cursor, N_NODES);
    gcn_csr_fill<<<gE, T, 0, stream>>>(src, dst, dinv, cursor, col, val);

    // layer 1: h = X@W1 -> bufA ; fused gather+self+bias+relu -> bufB
    gcn_gemm_in<<<gNH, T, 0, stream>>>(x, W1, bufA);
    gcn_gather<<<gGat, T, 0, stream>>>(rowptr, col, val, bufA, dinv, b1, bufB);

    // layer 2: h = bufB@W2 (fp32 WMMA) -> bufA ; fused gather -> bufB
    gcn_gemm_wmma<<<gWmma, T, 0, stream>>>(bufB, W2, bufA);
    gcn_gather<<<gGat, T, 0, stream>>>(rowptr, col, val, bufA, dinv, b2, bufB);

    // head
    gcn_head<<<gN, T, 0, stream>>>(bufB, Wout, bout, outp);
}